// NCELossMoco_83425444758003
// MI455X (gfx1250) — compile-verified
//
#include <hip/hip_runtime.h>
#include <math.h>

typedef __attribute__((ext_vector_type(2))) float v2f;
typedef __attribute__((ext_vector_type(8))) float v8f;

#define NROWS 2048
#define DIMC  128
#define KCOLS 65536
#define INVT  (1.0f/0.07f)      // 1/TEMPERATURE
#define CT    8                 // column tiles (of 16) per wave

// ---------------------------------------------------------------------------
// Stage 1: normalize rows, l_pos, per-row box params, zero rowsum accumulator
// ---------------------------------------------------------------------------
__device__ __forceinline__ float blockReduce128(float v, float* sdata) {
    const int t = threadIdx.x;
    __syncthreads();
    sdata[t] = v;
    __syncthreads();
    for (int s = 64; s > 0; s >>= 1) {
        if (t < s) sdata[t] += sdata[t + s];
        __syncthreads();
    }
    return sdata[0];
}

__global__ __launch_bounds__(128) void moco_stage1(
    const float* __restrict__ out0, const float* __restrict__ out1,
    const float* __restrict__ lwhz,
    float* __restrict__ n1, float* __restrict__ lpos,
    float* __restrict__ rl, float* __restrict__ rw, float* __restrict__ rh,
    float* __restrict__ rz, float* __restrict__ rv, float* __restrict__ rowsum)
{
    __shared__ float sdata[128];
    const int row = blockIdx.x;
    const int d   = threadIdx.x;

    const float x0 = out0[row * DIMC + d];
    const float x1 = out1[row * DIMC + d];

    const float s0 = blockReduce128(x0 * x0, sdata);
    const float s1 = blockReduce128(x1 * x1, sdata);
    const float i0 = 1.0f / fmaxf(sqrtf(s0), 1e-12f);
    const float i1 = 1.0f / fmaxf(sqrtf(s1), 1e-12f);
    const float a = x0 * i0;
    const float b = x1 * i1;
    n1[row * DIMC + d] = a;

    const float lp = blockReduce128(a * b, sdata);
    if (d == 0) {
        lpos[row] = lp;
        const float p0 = lwhz[row * 4 + 0];
        const float p1 = lwhz[row * 4 + 1];
        const float h  = lwhz[row * 4 + 2];
        const float z  = lwhz[row * 4 + 3];
        const float l  = fmaxf(p0, p1);
        const float w  = fminf(p0, p1);
        rl[row] = l; rw[row] = w; rh[row] = h; rz[row] = z; rv[row] = l * w * h;
        rowsum[row] = 0.0f;
    }
}

// ---------------------------------------------------------------------------
// Stage 2: fp32 WMMA GEMM tile + IoU weighting + exp accumulation per row
// ---------------------------------------------------------------------------
__global__ __launch_bounds__(256) void moco_stage2(
    const float* __restrict__ qf,     // (DIM, K) row-major
    const float* __restrict__ ql,     // (4, K) row-major
    const float* __restrict__ n1,     // (N, DIM)
    const float* __restrict__ rl, const float* __restrict__ rw,
    const float* __restrict__ rh, const float* __restrict__ rz,
    const float* __restrict__ rv,
    float* __restrict__ rowsum)
{
    __shared__ float s_rl[16], s_rw[16], s_rh[16], s_rz[16], s_rv[16];

    const int tid  = threadIdx.x;
    const int wave = tid >> 5;
    const int lane = tid & 31;
    const int m    = lane & 15;   // M (row within tile) / N (col within tile)
    const int kh   = lane >> 4;   // lane-half selects K pair {0,1} vs {2,3}
    const int row0 = blockIdx.x * 16;

    if (tid < 16) {
        s_rl[tid] = rl[row0 + tid];
        s_rw[tid] = rw[row0 + tid];
        s_rh[tid] = rh[row0 + tid];
        s_rz[tid] = rz[row0 + tid];
        s_rv[tid] = rv[row0 + tid];
    }
    __syncthreads();

    // --- load A tile (16 x 128 fp32) into registers, WMMA 16x16x4 A layout:
    // lane L<16: M=L holds K = 4*ks + {0,1}; lane L>=16: M=L-16 holds K = 4*ks + {2,3}
    v2f a[32];
    const float* arow = n1 + (size_t)(row0 + m) * DIMC + kh * 2;
#pragma unroll
    for (int ks = 0; ks < 32; ++ks)
        a[ks] = *reinterpret_cast<const v2f*>(arow + ks * 4);

    float acc[8];
#pragma unroll
    for (int j = 0; j < 8; ++j) acc[j] = 0.0f;

    const int c0 = (blockIdx.y * 8 + wave) * CT;

    for (int t = 0; t < CT; ++t) {
        const int col = (c0 + t) * 16 + m;

        v8f c = {};
        const float* bcol = qf + (size_t)(kh * 2) * KCOLS + col;
#pragma unroll
        for (int ks = 0; ks < 32; ++ks) {
            // B 4x16 layout mirrors A: v.x = row (4*ks+kh*2), v.y = row (4*ks+kh*2+1)
            v2f b;
            b.x = bcol[(size_t)(ks * 4 + 0) * KCOLS];
            b.y = bcol[(size_t)(ks * 4 + 1) * KCOLS];
            c = __builtin_amdgcn_wmma_f32_16x16x4_f32(
                    false, a[ks], false, b, (short)0, c, false, false);
        }

        // --- epilogue: IoU weighting + exp, per-column params uniform over j
        const float ln = ql[0 * (size_t)KCOLS + col];
        const float wn = ql[1 * (size_t)KCOLS + col];
        const float hn = ql[2 * (size_t)KCOLS + col];
        const float zn = ql[3 * (size_t)KCOLS + col];
        const float voln  = ln * wn * hn;
        const float zhi_n = zn + 0.5f * hn;
        const float zlo_n = zn - 0.5f * hn;

#pragma unroll
        for (int j = 0; j < 8; ++j) {
            const int rm = j + kh * 8;              // C layout: VGPR j, lane-half
            const float l = s_rl[rm], w = s_rw[rm], h = s_rh[rm];
            const float z = s_rz[rm], vol = s_rv[rm];
            const float mom = fminf(z + 0.5f * h, zhi_n);
            const float Mom = fmaxf(z - 0.5f * h, zlo_n);
            const float oh  = fmaxf(mom - Mom, 0.0f);
            const float ov  = fminf(l, ln) * fminf(w, wn) * oh;
            const float iou = ov / fmaxf(vol + voln - ov, 1e-6f);
            const float logit = c[j] * (1.0f - iou) * INVT;
            acc[j] += __expf(logit - INVT);          // shifted by fixed max 1/T
        }
    }

    // --- reduce each row's partial over its 16 lanes, 2 atomics per j
#pragma unroll
    for (int j = 0; j < 8; ++j) {
        float v = acc[j];
        v += __shfl_xor(v, 1, 32);
        v += __shfl_xor(v, 2, 32);
        v += __shfl_xor(v, 4, 32);
        v += __shfl_xor(v, 8, 32);
        if (m == 0) atomicAdd(&rowsum[row0 + j + kh * 8], v);
    }
}

// ---------------------------------------------------------------------------
// Stage 3: fold l_pos term, log, mean
// ---------------------------------------------------------------------------
__global__ __launch_bounds__(256) void moco_stage3(
    const float* __restrict__ rowsum, const float* __restrict__ lpos,
    float* __restrict__ out)
{
    __shared__ float sdata[256];
    const int t = threadIdx.x;
    float s = 0.0f;
    for (int r = t; r < NROWS; r += 256) {
        const float lp = lpos[r] * INVT;
        const float total = rowsum[r] + __expf(lp - INVT);
        s += (INVT + __logf(total)) - lp;   // LSE - lpos/T with shift M=1/T
    }
    sdata[t] = s;
    __syncthreads();
    for (int k = 128; k > 0; k >>= 1) {
        if (t < k) sdata[t] += sdata[t + k];
        __syncthreads();
    }
    if (t == 0) out[0] = sdata[0] * (1.0f / (float)NROWS);
}

// ---------------------------------------------------------------------------
extern "C" void kernel_launch(void* const* d_in, const int* in_sizes, int n_in,
                              void* d_out, int out_size, void* d_ws, size_t ws_size,
                              hipStream_t stream) {
    (void)in_sizes; (void)n_in; (void)out_size; (void)ws_size;
    const float* out0 = (const float*)d_in[0];   // (N, DIM)
    const float* out1 = (const float*)d_in[1];   // (N, DIM)
    const float* lwhz = (const float*)d_in[2];   // (N, 4)
    const float* qf   = (const float*)d_in[3];   // (DIM, K)
    const float* ql   = (const float*)d_in[4];   // (4, K)

    float* ws     = (float*)d_ws;
    float* n1     = ws;                       // N*DIM
    float* lpos   = n1   + (size_t)NROWS * DIMC;
    float* rl     = lpos + NROWS;
    float* rw     = rl   + NROWS;
    float* rh     = rw   + NROWS;
    float* rz     = rh   + NROWS;
    float* rv     = rz   + NROWS;
    float* rowsum = rv   + NROWS;

    moco_stage1<<<NROWS, 128, 0, stream>>>(out0, out1, lwhz,
                                           n1, lpos, rl, rw, rh, rz, rv, rowsum);

    dim3 grid(NROWS / 16, KCOLS / 16 / (8 * CT));   // 128 x 64
    moco_stage2<<<grid, 256, 0, stream>>>(qf, ql, n1, rl, rw, rh, rz, rv, rowsum);

    moco_stage3<<<1, 256, 0, stream>>>(rowsum, lpos, (float*)d_out);
}